// SoftSpearman_61641370632457
// MI455X (gfx1250) — compile-verified
//
#include <hip/hip_runtime.h>
#include <math.h>

typedef __attribute__((ext_vector_type(2))) float v2f;
typedef __attribute__((ext_vector_type(8))) float v8f;

#define N 512
#define NTHREADS 256

__device__ __forceinline__ float logaddexp_f(float a, float b) {
    float m = fmaxf(a, b);
    float d = fabsf(a - b);
    return m + log1pf(expf(-d));
}

// Sum a per-lane float across all 32 lanes of the wave using the FP32 WMMA.
// A (16x4 f32): lanes 0-15 supply M=lane {K0,K1}, lanes 16-31 M=lane-16 {K2,K3}.
// We load {v, 0}, so A[m][0]=v[m], A[m][2]=v[m+16], others 0. B = all-ones, so
// D[m][n] = v[m] + v[m+16] for every n. Sum the 8 C/D VGPRs (rows 0-7 on lanes
// 0-15, rows 8-15 on lanes 16-31) and fold with one shfl_xor(16).
__device__ __forceinline__ float wmma_wave_sum(float v) {
    v2f a; a.x = v;    a.y = 0.0f;
    v2f b; b.x = 1.0f; b.y = 1.0f;
    v8f c = {};
    c = __builtin_amdgcn_wmma_f32_16x16x4_f32(
        /*neg_a=*/false, a, /*neg_b=*/false, b,
        /*c_mod=*/(short)0, c, /*reuse_a=*/false, /*reuse_b=*/false);
    float t = c[0] + c[1] + c[2] + c[3] + c[4] + c[5] + c[6] + c[7];
    t += __shfl_xor(t, 16, 32);
    return t;  // full 32-lane sum in every lane
}

// Compute KL-regularized soft ranks for one length-512 row into ranks[0..511]
// (original element order). Scratch arrays are LDS.
__device__ void soft_rank_row(const float* __restrict__ src,
                              float* __restrict__ ranks,
                              float* skey, int* sidx,
                              float* blkS, float* blkW, float* blkV,
                              int* blkStart, float* dual) {
    const int tid = threadIdx.x;

    // load + scale by 1/reg_strength = 10
    for (int t = tid; t < N; t += NTHREADS) {
        skey[t] = src[t] * 10.0f;
        sidx[t] = t;
    }
    __syncthreads();

    // in-LDS bitonic sort, descending, with index payload
    for (unsigned k = 2; k <= (unsigned)N; k <<= 1) {
        for (unsigned j = k >> 1; j > 0; j >>= 1) {
            for (unsigned t = tid; t < (unsigned)N; t += NTHREADS) {
                unsigned ixj = t ^ j;
                if (ixj > t) {
                    float a = skey[t], bb = skey[ixj];
                    bool up = ((t & k) == 0);
                    if (up ? (a < bb) : (a > bb)) {
                        skey[t] = bb; skey[ixj] = a;
                        int ia = sidx[t]; sidx[t] = sidx[ixj]; sidx[ixj] = ia;
                    }
                }
            }
            __syncthreads();
        }
    }

    // PAV (pool adjacent violators) for non-increasing KL-isotonic fit.
    // Block state: lse of s over block (blkS), sum of exp(w)=integer weights
    // (blkW, exact), block value v = blkS - log(blkW), start index.
    if (tid == 0) {
        int top = -1;
        for (int i = 0; i < N; ++i) {
            float ls = skey[i];
            float sw = (float)(N - i);   // w_i = log(n - i) at sorted pos i
            int start = i;
            float v = ls - logf(sw);
            while (top >= 0 && v > blkV[top]) {   // violation -> merge
                ls = logaddexp_f(ls, blkS[top]);
                sw += blkW[top];
                start = blkStart[top];
                --top;
                v = ls - logf(sw);
            }
            ++top;
            blkS[top] = ls; blkW[top] = sw; blkV[top] = v; blkStart[top] = start;
        }
        // broadcast block values to per-element dual
        for (int bI = 0; bI <= top; ++bI) {
            int s0 = blkStart[bI];
            int s1 = (bI == top) ? N : blkStart[bI + 1];
            float v = blkV[bI];
            for (int i = s0; i < s1; ++i) dual[i] = v;
        }
    }
    __syncthreads();

    // ranks_sorted = exp(s - dual); scatter back to original order
    for (int t = tid; t < N; t += NTHREADS) {
        ranks[sidx[t]] = expf(skey[t] - dual[t]);
    }
    __syncthreads();
}

__global__ __launch_bounds__(NTHREADS)
void SoftSpearman_kernel(const float* __restrict__ x1,
                         const float* __restrict__ x2,
                         float* __restrict__ out) {
    __shared__ float skey[N];
    __shared__ int   sidx[N];
    __shared__ float blkS[N];
    __shared__ float blkW[N];
    __shared__ float blkV[N];
    __shared__ int   blkStart[N];
    __shared__ float dual[N];
    __shared__ float ranks1[N];
    __shared__ float ranks2[N];

    const int row = blockIdx.x;

    soft_rank_row(x1 + (size_t)row * N, ranks1,
                  skey, sidx, blkS, blkW, blkV, blkStart, dual);
    soft_rank_row(x2 + (size_t)row * N, ranks2,
                  skey, sidx, blkS, blkW, blkV, blkStart, dual);

    // Pearson correlation of the two rank vectors: wave 0 only (EXEC all-ones
    // inside the wave — required for WMMA).
    if (threadIdx.x < 32) {
        const int lane = threadIdx.x;
        float s1 = 0.0f, s2 = 0.0f;
        for (int i = lane; i < N; i += 32) { s1 += ranks1[i]; s2 += ranks2[i]; }
        s1 = wmma_wave_sum(s1);
        s2 = wmma_wave_sum(s2);
        const float m1 = s1 * (1.0f / N);
        const float m2 = s2 * (1.0f / N);

        float c11 = 0.0f, c22 = 0.0f, c12 = 0.0f;
        for (int i = lane; i < N; i += 32) {
            float a = ranks1[i] - m1;
            float b = ranks2[i] - m2;
            c11 += a * a; c22 += b * b; c12 += a * b;
        }
        c11 = wmma_wave_sum(c11);
        c22 = wmma_wave_sum(c22);
        c12 = wmma_wave_sum(c12);

        if (lane == 0) {
            float corr = c12 / (sqrtf(c11) * sqrtf(c22));
            out[row] = 1.0f - corr;
        }
    }
}

extern "C" void kernel_launch(void* const* d_in, const int* in_sizes, int n_in,
                              void* d_out, int out_size, void* d_ws, size_t ws_size,
                              hipStream_t stream) {
    const float* x1 = (const float*)d_in[0];
    const float* x2 = (const float*)d_in[1];
    float* out = (float*)d_out;
    const int rows = in_sizes[0] / N;  // 8*8 = 64 rows
    SoftSpearman_kernel<<<rows, NTHREADS, 0, stream>>>(x1, x2, out);
}